// PQC2_60833916781187
// MI455X (gfx1250) — compile-verified
//
#include <hip/hip_runtime.h>

typedef float v2f __attribute__((ext_vector_type(2)));
typedef float v8f __attribute__((ext_vector_type(8)));

#define NQ     14
#define DIMS   (1 << NQ)        // 16384 amplitudes
#define BATCH  256
#define LAYERS 3
#define NPAIRS (NQ - 1)         // 13 ZZ pairs (range-1)
#define NMAT   (LAYERS * 2 * 4) // 24 fused 16x16 group unitaries

__device__ __forceinline__ void cmul(float ar, float ai, float br, float bi,
                                     float& cr, float& ci) {
    cr = ar * br - ai * bi;
    ci = ar * bi + ai * br;
}

// ---------------------------------------------------------------------------
// Prep kernel: build the 24 fused 16x16 complex unitaries (kron of four 2x2
// rotation gates; group 3 covers bits 10..13 with identity on bits 10,11).
// One block per matrix, one thread per (row a, col k) entry.
// ---------------------------------------------------------------------------
__global__ void pqc_prep_unitaries(const float* __restrict__ psq, // [L][2][NQ][3]
                                   float* __restrict__ Ur,
                                   float* __restrict__ Ui) {
    const int mid = blockIdx.x;   // (l*2+r)*4+g
    const int g   = mid & 3;
    const int lr  = mid >> 2;     // l*2 + r
    const int tid = threadIdx.x;  // 0..255
    const int a = tid >> 4, k = tid & 15;
    const int b = (g < 3) ? g * 4 : 10;

    float accr = 1.f, acci = 0.f;
    for (int j = 0; j < 4; ++j) {
        const int bp = b + j;  // global bit position
        float g00r, g00i, g01r, g01i, g10r, g10i, g11r, g11i;
        if (g == 3 && j < 2) {  // identity on bits 10,11 in last group
            g00r = 1.f; g00i = 0.f; g01r = 0.f; g01i = 0.f;
            g10r = 0.f; g10i = 0.f; g11r = 1.f; g11i = 0.f;
        } else {
            const int q = NQ - 1 - bp;  // qubit index for this bit
            const float* ang = psq + (lr * NQ + q) * 3;
            const float phi = ang[0], th = ang[1], om = ang[2];
            const float c = cosf(0.5f * th), s = sinf(0.5f * th);
            const float app = 0.5f * (phi + om), amm = 0.5f * (phi - om);
            const float epr = cosf(app), epi = -sinf(app);  // e^{-i(phi+om)/2}
            const float emr = cosf(amm), emi = -sinf(amm);  // e^{-i(phi-om)/2}
            g00r =  epr * c;  g00i =  epi * c;   // ep*c
            g01r = -emr * s;  g01i =  emi * s;   // -conj(em)*s
            g10r =  emr * s;  g10i =  emi * s;   // em*s
            g11r =  epr * c;  g11i = -epi * c;   // conj(ep)*c
        }
        const int ab = (a >> j) & 1, kb = (k >> j) & 1;
        float fr, fi;
        if      (ab == 0 && kb == 0) { fr = g00r; fi = g00i; }
        else if (ab == 0)            { fr = g01r; fi = g01i; }
        else if (kb == 0)            { fr = g10r; fi = g10i; }
        else                         { fr = g11r; fi = g11i; }
        float nr, ni;
        cmul(accr, acci, fr, fi, nr, ni);
        accr = nr; acci = ni;
    }
    Ur[mid * 256 + a * 16 + k] = accr;
    Ui[mid * 256 + a * 16 + k] = acci;
}

// ---------------------------------------------------------------------------
// Main kernel: one block per batch element; entire 16384-amplitude state held
// in LDS (128 KB of the WGP's 320 KB). Gate groups applied as complex 16x16
// GEMMs via V_WMMA_F32_16X16X4_F32 (4 real WMMA chains per complex product).
// ---------------------------------------------------------------------------
__global__ __launch_bounds__(256)
void pqc_main(const float* __restrict__ x,     // [B][2*NQ]
              const float* __restrict__ ptq,   // [L][NQ*NQ]
              const float* __restrict__ penc,  // [L][NQ*NQ]
              const float* __restrict__ pcl,   // [NQ]
              const float* __restrict__ Ur,
              const float* __restrict__ Ui,
              float* __restrict__ out) {
    __shared__ float sR[DIMS];
    __shared__ float sI[DIMS];
    __shared__ float sTheta[LAYERS * NPAIRS];
    __shared__ float sRed[256];

    const int n    = blockIdx.x;
    const int tid  = threadIdx.x;
    const int lane = tid & 31;
    const int wave = tid >> 5;

    // |0...0> initial state
    for (int t = tid; t < DIMS; t += 256) { sR[t] = 0.f; sI[t] = 0.f; }
    if (tid == 0) sR[0] = 1.f;

    // per-block ZZ angles: theta[l][p] = ptq + penc * <x2_i, x2_j>
    if (tid < LAYERS * NPAIRS) {
        const int l = tid / NPAIRS, p = tid % NPAIRS;
        const int i = p, j = p + 1;
        const int pf = i * NQ + j;
        const float f = x[n * 2 * NQ + 2 * i] * x[n * 2 * NQ + 2 * j] +
                        x[n * 2 * NQ + 2 * i + 1] * x[n * 2 * NQ + 2 * j + 1];
        sTheta[tid] = ptq[l * NQ * NQ + pf] + penc[l * NQ * NQ + pf] * f;
    }
    __syncthreads();

    for (int l = 0; l < LAYERS; ++l) {
        for (int r = 0; r < 2; ++r) {
            if (r == 1) {
                // diagonal ZZ phase: amp *= exp(-0.5i * sum_p theta_p * zz_p(idx))
                for (int t = 0; t < DIMS / 256; ++t) {
                    const int idx = tid + 256 * t;
                    float ang = 0.f;
#pragma unroll
                    for (int p = 0; p < NPAIRS; ++p) {
                        const int sgn = ((idx >> (NQ - 1 - p)) ^ (idx >> (NQ - 2 - p))) & 1;
                        const float th = sTheta[l * NPAIRS + p];
                        ang += sgn ? -th : th;
                    }
                    const float ca =  cosf(0.5f * ang);
                    const float sa = -sinf(0.5f * ang);
                    const float rr = sR[idx], ii = sI[idx];
                    sR[idx] = rr * ca - ii * sa;
                    sI[idx] = rr * sa + ii * ca;
                }
                __syncthreads();
            }
            for (int g = 0; g < 4; ++g) {
                const int b   = (g < 3) ? g * 4 : 10;
                const int mid = (l * 2 + r) * 4 + g;
                const float* ur = Ur + mid * 256;
                const float* ui = Ui + mid * 256;

                // A-operand (16x4 f32 chunks): lanes 0-15 hold rows M=0..15,
                // VGPR slot/half hold K.  Pre-negated imag copy for Cr term.
                const int arow = lane & 15;
                const int hi   = lane >> 4;
                v2f aR[4], aI[4], aN[4];
#pragma unroll
                for (int kc = 0; kc < 4; ++kc) {
                    const int k = kc * 4 + hi * 2;
                    aR[kc].x = ur[arow * 16 + k];
                    aR[kc].y = ur[arow * 16 + k + 1];
                    aI[kc].x = ui[arow * 16 + k];
                    aI[kc].y = ui[arow * 16 + k + 1];
                    aN[kc].x = -aI[kc].x;
                    aN[kc].y = -aI[kc].y;
                }

                const int nn = lane & 15;  // this lane's N column
                // 64 tiles of (16 amps x 16 groups); 8 tiles per wave
                for (int tt = 0; tt < 8; ++tt) {
                    const int tile = wave * 8 + tt;
                    const int G    = tile * 16 + nn;  // group id for this lane's column
                    const int base = (G & ((1 << b) - 1)) | ((G >> b) << (b + 4));
                    v8f dR = {0.f, 0.f, 0.f, 0.f, 0.f, 0.f, 0.f, 0.f};
                    v8f dI = {0.f, 0.f, 0.f, 0.f, 0.f, 0.f, 0.f, 0.f};
#pragma unroll
                    for (int kc = 0; kc < 4; ++kc) {
                        const int k  = kc * 4 + hi * 2;
                        const int i0 = base + (k << b);
                        const int i1 = base + ((k + 1) << b);
                        v2f bR, bI;
                        bR.x = sR[i0]; bR.y = sR[i1];
                        bI.x = sI[i0]; bI.y = sI[i1];
                        // Cr += Ar*Br - Ai*Bi ; Ci += Ar*Bi + Ai*Br
                        dR = __builtin_amdgcn_wmma_f32_16x16x4_f32(false, aR[kc], false, bR, (short)0, dR, false, false);
                        dR = __builtin_amdgcn_wmma_f32_16x16x4_f32(false, aN[kc], false, bI, (short)0, dR, false, false);
                        dI = __builtin_amdgcn_wmma_f32_16x16x4_f32(false, aR[kc], false, bI, (short)0, dI, false, false);
                        dI = __builtin_amdgcn_wmma_f32_16x16x4_f32(false, aI[kc], false, bR, (short)0, dI, false, false);
                    }
                    // D layout: VGPR r -> M=r (lanes 0-15) / M=r+8 (lanes 16-31), N=lane&15
#pragma unroll
                    for (int r8 = 0; r8 < 8; ++r8) {
                        const int m   = r8 + hi * 8;
                        const int idx = base + (m << b);
                        sR[idx] = dR[r8];
                        sI[idx] = dI[r8];
                    }
                }
                __syncthreads();
            }
        }
    }

    // ----- observables, fused with classical coefficients -----
    float pc[NQ];
#pragma unroll
    for (int q = 0; q < NQ; ++q) pc[q] = pcl[q];

    float acc = 0.f;
    // sum_q pc[q] * <Z_q>  : per-amplitude weighted probability
    for (int t = 0; t < DIMS / 256; ++t) {
        const int idx = tid + 256 * t;
        const float rr = sR[idx], ii = sI[idx];
        const float prob = rr * rr + ii * ii;
        float wz = 0.f;
#pragma unroll
        for (int q = 0; q < NQ; ++q) {
            const int bit = (idx >> (NQ - 1 - q)) & 1;
            wz += bit ? -pc[q] : pc[q];
        }
        acc += prob * wz;
    }
    // sum_q pc[13-q] * <X_q> ; qubit q pairs along bit p = 13-q, weight = pc[p]
    for (int p = 0; p < NQ; ++p) {
        const float cx = 2.f * pc[p];
        float xacc = 0.f;
        for (int t = 0; t < (DIMS / 2) / 256; ++t) {
            const int m  = tid + 256 * t;
            const int i0 = (m & ((1 << p) - 1)) | ((m >> p) << (p + 1));
            const int i1 = i0 | (1 << p);
            xacc += sR[i0] * sR[i1] + sI[i0] * sI[i1];
        }
        acc += cx * xacc;
    }

    sRed[tid] = acc;
    __syncthreads();
    for (int s = 128; s > 0; s >>= 1) {
        if (tid < s) sRed[tid] += sRed[tid + s];
        __syncthreads();
    }
    if (tid == 0) out[n] = sRed[0];
}

// ---------------------------------------------------------------------------
extern "C" void kernel_launch(void* const* d_in, const int* in_sizes, int n_in,
                              void* d_out, int out_size, void* d_ws, size_t ws_size,
                              hipStream_t stream) {
    const float* x    = (const float*)d_in[0];  // [B, 28]
    const float* psq  = (const float*)d_in[1];  // [L, 2, NQ, 3]
    const float* ptq  = (const float*)d_in[2];  // [L, NQ*NQ]
    const float* penc = (const float*)d_in[3];  // [L, NQ*NQ]
    const float* pcl  = (const float*)d_in[4];  // [NQ]
    float* out = (float*)d_out;                 // [B]

    float* Ur = (float*)d_ws;                   // 24*256 floats
    float* Ui = Ur + NMAT * 256;                // 24*256 floats

    pqc_prep_unitaries<<<NMAT, 256, 0, stream>>>(psq, Ur, Ui);
    pqc_main<<<BATCH, 256, 0, stream>>>(x, ptq, penc, pcl, Ur, Ui, out);
}